// Classifier_9783935500741
// MI455X (gfx1250) — compile-verified
//
#include <hip/hip_runtime.h>
#include <hip/hip_bf16.h>

#define N_NODES 100000
#define N_EDGES 1600000
#define DIM      64
#define NGRAPH   128
#define NCLASS   20

typedef __attribute__((ext_vector_type(2))) float v2f;
typedef __attribute__((ext_vector_type(8))) float v8f;

// ---------------------------------------------------------------- utilities
__global__ void zero_kernel(float* __restrict__ p, long n) {
  long i = (long)blockIdx.x * blockDim.x + threadIdx.x;
  if (i < n) p[i] = 0.0f;
}

// h[n][0:64] = emb_table[tokens[n]][0:64]   (float4 per thread, 16 threads/row)
__global__ void embed_kernel(const int* __restrict__ tok,
                             const float* __restrict__ emb,
                             float* __restrict__ h, long total) {
  long t = (long)blockIdx.x * blockDim.x + threadIdx.x;
  if (t >= total) return;
  long n  = t >> 4;
  int  d4 = (int)(t & 15) << 2;
  const float4 v = *(const float4*)(emb + (size_t)tok[n] * DIM + d4);
  *(float4*)(h + (size_t)n * DIM + d4) = v;
}

// deg[dst[e]] += 1.0f
__global__ void degree_kernel(const int* __restrict__ dst,
                              float* __restrict__ deg, int e_cnt) {
  int e = blockIdx.x * blockDim.x + threadIdx.x;
  if (e < e_cnt) atomicAdd(deg + dst[e], 1.0f);
}

// deg -> 1/max(deg,1)  (in place)
__global__ void invdeg_kernel(float* __restrict__ deg, int n_cnt) {
  int i = blockIdx.x * blockDim.x + threadIdx.x;
  if (i < n_cnt) deg[i] = 1.0f / fmaxf(deg[i], 1.0f);
}

// agg[dst[e]][:] += h[src[e]][:]   16 threads/edge, float4 each (coalesced)
__global__ void scatter_kernel(const int* __restrict__ src,
                               const int* __restrict__ dst,
                               const float* __restrict__ h,
                               float* __restrict__ agg, long total) {
  long t = (long)blockIdx.x * blockDim.x + threadIdx.x;
  if (t >= total) return;
  long e  = t >> 4;
  int  d4 = (int)(t & 15) << 2;
  int  s  = src[e];
  int  d  = dst[e];
  const float4 v = *(const float4*)(h + (size_t)s * DIM + d4);
  float* o = agg + (size_t)d * DIM + d4;
  atomicAdd(o + 0, v.x);
  atomicAdd(o + 1, v.y);
  atomicAdd(o + 2, v.z);
  atomicAdd(o + 3, v.w);
}

// Y = relu((X * inv_deg_row) @ W + b), W is 64x64 staged in LDS.
// One wave32 per 16x16 output tile; V_WMMA_F32_16X16X4_F32, 16 k-steps.
// block = (32, 8): 8 waves/block share the LDS copy of W.
__global__ void gcn_gemm_relu(const float* __restrict__ X,
                              const float* __restrict__ inv_deg,
                              const float* __restrict__ W,
                              const float* __restrict__ bias,
                              float* __restrict__ Y, int nrows) {
  __shared__ float Ws[DIM * DIM];  // 16 KB
  __shared__ float bs[DIM];
  int t = threadIdx.y * 32 + threadIdx.x;  // 0..255
#pragma unroll
  for (int i = 0; i < 16; ++i) Ws[t * 16 + i] = W[t * 16 + i];
  if (t < DIM) bs[t] = bias[t];
  __syncthreads();

  int waveId  = blockIdx.x * blockDim.y + threadIdx.y;
  int row     = (waveId >> 2) * 16;   // row tile (nrows/16 tiles)
  int col     = (waveId & 3) * 16;    // 4 col tiles of 16 (DIM=64)
  if (row >= nrows) return;           // wave-uniform (EXEC stays full)

  int lane = threadIdx.x;
  int m    = lane & 15;
  int half = lane >> 4;               // 0: K={0,1}, 1: K={2,3} per 4-step

  const float  scale = inv_deg[row + m];
  const float* xrow  = X + (size_t)(row + m) * DIM;

  v8f c = {};
#pragma unroll
  for (int k = 0; k < DIM; k += 4) {
    int ka = k + half * 2;
    v2f a, b;
    // A 16x4 f32 layout: lane m holds A[m][ka], A[m][ka+1] in 2 VGPRs
    a.x = xrow[ka + 0] * scale;
    a.y = xrow[ka + 1] * scale;
    // B 4x16 f32 layout: VGPR r, half h -> K = r + 2*h, N = lane&15
    b.x = Ws[(ka + 0) * DIM + col + m];
    b.y = Ws[(ka + 1) * DIM + col + m];
    c = __builtin_amdgcn_wmma_f32_16x16x4_f32(false, a, false, b,
                                              (short)0, c, false, false);
  }

  const float bval = bs[col + m];
#pragma unroll
  for (int r = 0; r < 8; ++r) {
    // C/D layout: VGPR r -> M = r + 8*half, N = lane&15
    int   mo = row + r + half * 8;
    float v  = c[r] + bval;
    v = v > 0.0f ? v : 0.0f;
    Y[(size_t)mo * DIM + col + m] = v;
  }
}

// cnt[gid[n]] += 1
__global__ void graph_count_kernel(const int* __restrict__ gid,
                                   float* __restrict__ cnt, int n_cnt) {
  int i = blockIdx.x * blockDim.x + threadIdx.x;
  if (i < n_cnt) atomicAdd(cnt + gid[i], 1.0f);
}

// pool[gid[n]][:] += h[n][:]
__global__ void pool_scatter_kernel(const int* __restrict__ gid,
                                    const float* __restrict__ h,
                                    float* __restrict__ pool, long total) {
  long t = (long)blockIdx.x * blockDim.x + threadIdx.x;
  if (t >= total) return;
  long n  = t >> 4;
  int  d4 = (int)(t & 15) << 2;
  int  g  = gid[n];
  const float4 v = *(const float4*)(h + (size_t)n * DIM + d4);
  float* o = pool + (size_t)g * DIM + d4;
  atomicAdd(o + 0, v.x);
  atomicAdd(o + 1, v.y);
  atomicAdd(o + 2, v.z);
  atomicAdd(o + 3, v.w);
}

// out[g][c] = (pool[g]/max(cnt[g],1)) @ Wc + bc   (G*C = 2560 threads)
__global__ void classify_kernel(const float* __restrict__ pool,
                                const float* __restrict__ cnt,
                                const float* __restrict__ Wc,
                                const float* __restrict__ bc,
                                float* __restrict__ out) {
  int t = blockIdx.x * blockDim.x + threadIdx.x;
  if (t >= NGRAPH * NCLASS) return;
  int g = t / NCLASS;
  int c = t % NCLASS;
  float inv = 1.0f / fmaxf(cnt[g], 1.0f);
  float acc = bc[c];
  const float* pg = pool + (size_t)g * DIM;
#pragma unroll
  for (int d = 0; d < DIM; ++d) acc += pg[d] * inv * Wc[d * NCLASS + c];
  out[t] = acc;
}

// ---------------------------------------------------------------- launch
extern "C" void kernel_launch(void* const* d_in, const int* in_sizes, int n_in,
                              void* d_out, int out_size, void* d_ws, size_t ws_size,
                              hipStream_t stream) {
  const int*   tokens    = (const int*)d_in[0];
  const int*   edge_src  = (const int*)d_in[1];
  const int*   edge_dst  = (const int*)d_in[2];
  const int*   graph_ids = (const int*)d_in[3];
  const float* emb_table = (const float*)d_in[4];
  const float* W1        = (const float*)d_in[5];
  const float* b1        = (const float*)d_in[6];
  const float* W2        = (const float*)d_in[7];
  const float* b2        = (const float*)d_in[8];
  const float* Wc        = (const float*)d_in[9];
  const float* bc        = (const float*)d_in[10];
  float*       out       = (float*)d_out;

  // workspace layout (floats)
  float* hA      = (float*)d_ws;                       // N*D
  float* hB      = hA + (size_t)N_NODES * DIM;         // N*D (agg)
  float* inv_deg = hB + (size_t)N_NODES * DIM;         // N
  float* pool    = inv_deg + N_NODES;                  // G*D
  float* cnt     = pool + (size_t)NGRAPH * DIM;        // G

  const int BS = 256;
  const long zeroN = (long)N_NODES * DIM + N_NODES + NGRAPH * DIM + NGRAPH;

  // 1. zero agg + deg + pool + cnt (hA fully overwritten by embed)
  zero_kernel<<<(int)((zeroN + BS - 1) / BS), BS, 0, stream>>>(hB, zeroN);

  // 2. embedding gather
  const long embT = (long)N_NODES * 16;
  embed_kernel<<<(int)(embT / BS), BS, 0, stream>>>(tokens, emb_table, hA, embT);

  // 3. degree + inv_deg
  degree_kernel<<<N_EDGES / BS, BS, 0, stream>>>(edge_dst, inv_deg, N_EDGES);
  invdeg_kernel<<<(N_NODES + BS - 1) / BS, BS, 0, stream>>>(inv_deg, N_NODES);

  const long scatT  = (long)N_EDGES * 16;
  const int  scatB  = (int)(scatT / BS);
  const int  tiles  = (N_NODES / 16) * 4;   // 25000 wave-tiles
  dim3 gblock(32, 8);
  const int  gemmG  = tiles / 8;            // 3125 blocks

  // 4. layer 1: scatter hA -> hB, gemm hB -> hA
  scatter_kernel<<<scatB, BS, 0, stream>>>(edge_src, edge_dst, hA, hB, scatT);
  gcn_gemm_relu<<<gemmG, gblock, 0, stream>>>(hB, inv_deg, W1, b1, hA, N_NODES);

  // 5. layer 2: re-zero agg, scatter hA -> hB, gemm hB -> hA
  const long aggN = (long)N_NODES * DIM;
  zero_kernel<<<(int)((aggN + BS - 1) / BS), BS, 0, stream>>>(hB, aggN);
  scatter_kernel<<<scatB, BS, 0, stream>>>(edge_src, edge_dst, hA, hB, scatT);
  gcn_gemm_relu<<<gemmG, gblock, 0, stream>>>(hB, inv_deg, W2, b2, hA, N_NODES);

  // 6. per-graph mean pool + classifier
  graph_count_kernel<<<(N_NODES + BS - 1) / BS, BS, 0, stream>>>(graph_ids, cnt, N_NODES);
  pool_scatter_kernel<<<(int)(embT / BS), BS, 0, stream>>>(graph_ids, hA, pool, embT);
  classify_kernel<<<(NGRAPH * NCLASS + BS - 1) / BS, BS, 0, stream>>>(pool, cnt, Wc, bc, out);
}